// PMLP_with_EdgeAttr_60936995996176
// MI455X (gfx1250) — compile-verified
//
#include <hip/hip_runtime.h>
#include <hip/hip_bf16.h>

typedef __attribute__((ext_vector_type(2))) float v2f;
typedef __attribute__((ext_vector_type(8))) float v8f;

#define HF 128  // feature width (IN == H == OUT == EA == 128)
#define MTILE 64  // rows per block in the WMMA GEMM

// ---------------------------------------------------------------------------
// init: agg_ea = 0, deg = 1 (self loop)
// ---------------------------------------------------------------------------
__global__ void init_nodes(float* __restrict__ agg_ea, float* __restrict__ deg,
                           long long n128, int n) {
  long long t = (long long)blockIdx.x * blockDim.x + threadIdx.x;
  if (t < n128) agg_ea[t] = 0.0f;
  if (t < n)    deg[t]    = 1.0f;
}

// ---------------------------------------------------------------------------
// deg[dst[e]] += 1
// ---------------------------------------------------------------------------
__global__ void deg_accum(const int* __restrict__ dst, float* __restrict__ deg, int nE) {
  int e = blockIdx.x * blockDim.x + threadIdx.x;
  if (e < nE) atomicAdd(&deg[dst[e]], 1.0f);
}

// ---------------------------------------------------------------------------
// deg_inv = 1/deg ; bn_scale/bn_shift precompute
// ---------------------------------------------------------------------------
__global__ void finalize_stats(const float* __restrict__ deg, float* __restrict__ dinv, int n,
                               const float* __restrict__ gamma, const float* __restrict__ beta,
                               const float* __restrict__ mean,  const float* __restrict__ var,
                               float* __restrict__ bns, float* __restrict__ bnb) {
  int t = blockIdx.x * blockDim.x + threadIdx.x;
  if (t < n) dinv[t] = 1.0f / deg[t];
  if (t < HF) {
    float s = gamma[t] * rsqrtf(var[t] + 1e-5f);
    bns[t] = s;
    bnb[t] = beta[t] - mean[t] * s;
  }
}

// ---------------------------------------------------------------------------
// out[didx[e]] += data[row(e)], 128-float rows. One wave per edge, float4/lane.
// sidx==nullptr -> row = e (edge_attr scatter); else row = sidx[e] (h gather).
// data rows are L2-resident (51 MB << 192 MB L2); atomics resolve at L2.
// ---------------------------------------------------------------------------
__global__ void scatter_rows(const float* __restrict__ data,
                             const int* __restrict__ sidx,
                             const int* __restrict__ didx,
                             float* __restrict__ out, int nE) {
  int e = blockIdx.x * 8 + (threadIdx.x >> 5);
  if (e >= nE) return;
  int lane = threadIdx.x & 31;
  int srow = sidx ? sidx[e] : e;
  int drow = didx[e];
  const float4 v = *(const float4*)(data + (size_t)srow * HF + lane * 4);
  float* o = out + (size_t)drow * HF + lane * 4;
  atomicAdd(o + 0, v.x);
  atomicAdd(o + 1, v.y);
  atomicAdd(o + 2, v.z);
  atomicAdd(o + 3, v.w);
}

// ---------------------------------------------------------------------------
// Fused GEMM: out0 = epi( A[nrows x 128] @ W[:, woff:woff+128]^T )
//   epi: *dinv[row], +addC[row,col], +bias[col], *bns+bnb, tanh  (all optional)
// Block: 256 thr = 8 waves; block owns 64 rows x 128 cols. Wave w owns cols
// [16w,16w+16) across 4 row tiles -> each B fragment feeds 4 WMMAs.
// A tile (64x128, 32 KB) staged in LDS; 128 x V_WMMA_F32_16X16X4_F32 per wave.
// ---------------------------------------------------------------------------
__global__ __launch_bounds__(256)
void gemm128_wmma(const float* __restrict__ A, const float* __restrict__ W,
                  int ldw, int woff,
                  const float* __restrict__ dinv, const float* __restrict__ addC,
                  const float* __restrict__ bias,
                  const float* __restrict__ bns, const float* __restrict__ bnb,
                  int do_tanh,
                  float* __restrict__ out0, float* __restrict__ out1, int nrows) {
  __shared__ float As[MTILE * HF];

  const int m0 = blockIdx.x * MTILE;

  // cooperative A-tile load: 64*128 floats = 2048 float4 = 256 thr x 8
  {
    int t = threadIdx.x;
    #pragma unroll
    for (int i = 0; i < 8; ++i) {
      int idx = t + i * 256;     // float4 index
      int row = idx >> 5;        // 32 float4 per row
      int k4  = idx & 31;
      float4 v = make_float4(0.f, 0.f, 0.f, 0.f);
      int grow = m0 + row;
      if (grow < nrows) v = *(const float4*)(A + (size_t)grow * HF + k4 * 4);
      *(float4*)(As + row * HF + k4 * 4) = v;
    }
  }
  __syncthreads();

  const int wave = threadIdx.x >> 5;   // 0..7 -> col tile
  const int lane = threadIdx.x & 31;
  const int lm   = lane & 15;
  const int half = lane >> 4;          // K sub-pair select (K+0/1 vs K+2/3)
  const int c0   = wave * 16;

  v8f acc[4] = {v8f{}, v8f{}, v8f{}, v8f{}};
  const float* abase = As + lm * HF + half * 2;
  const float* wbase = W + (size_t)(c0 + lm) * ldw + woff + half * 2;

  #pragma unroll
  for (int k0 = 0; k0 < HF; k0 += 4) {
    v2f b  = *(const v2f*)(wbase + k0);             // global_load_b64, reused 4x
    v2f a0 = *(const v2f*)(abase + k0);             // ds_load_b64
    v2f a1 = *(const v2f*)(abase + 16 * HF + k0);
    v2f a2 = *(const v2f*)(abase + 32 * HF + k0);
    v2f a3 = *(const v2f*)(abase + 48 * HF + k0);
    acc[0] = __builtin_amdgcn_wmma_f32_16x16x4_f32(false, a0, false, b, (short)0, acc[0], false, false);
    acc[1] = __builtin_amdgcn_wmma_f32_16x16x4_f32(false, a1, false, b, (short)0, acc[1], false, false);
    acc[2] = __builtin_amdgcn_wmma_f32_16x16x4_f32(false, a2, false, b, (short)0, acc[2], false, false);
    acc[3] = __builtin_amdgcn_wmma_f32_16x16x4_f32(false, a3, false, b, (short)0, acc[3], false, false);
  }

  // epilogue (C/D layout: VGPR r, lane L -> row = rowtile + r + 8*half, col = c0+(L&15))
  const int col  = c0 + lm;
  const float bcol = bias ? bias[col] : 0.0f;
  const float bs   = bns  ? bns[col]  : 1.0f;
  const float bb   = bnb  ? bnb[col]  : 0.0f;
  #pragma unroll
  for (int rt = 0; rt < 4; ++rt) {
    #pragma unroll
    for (int r = 0; r < 8; ++r) {
      int row = m0 + rt * 16 + r + half * 8;
      if (row < nrows) {
        float v = acc[rt][r];
        if (dinv) v *= dinv[row];
        if (addC) v += addC[(size_t)row * HF + col];
        v += bcol;
        if (bns) v = v * bs + bb;
        if (do_tanh) v = tanhf(v);
        size_t o = (size_t)row * HF + col;
        out0[o] = v;
        if (out1) out1[o] = v;
      }
    }
  }
}

// ---------------------------------------------------------------------------
// Host launcher
// ---------------------------------------------------------------------------
extern "C" void kernel_launch(void* const* d_in, const int* in_sizes, int n_in,
                              void* d_out, int out_size, void* d_ws, size_t ws_size,
                              hipStream_t stream) {
  const float* x    = (const float*)d_in[0];
  const int*   ei   = (const int*)  d_in[1];   // [2, E]
  const float* ea   = (const float*)d_in[2];   // [E, 128]
  const float* Wl[4] = { (const float*)d_in[3], (const float*)d_in[5],
                         (const float*)d_in[7], (const float*)d_in[9] };
  const float* bl[4] = { (const float*)d_in[4], (const float*)d_in[6],
                         (const float*)d_in[8], (const float*)d_in[10] };
  const float* Wc   = (const float*)d_in[11];  // [128, 256]
  const float* bc   = (const float*)d_in[12];
  const float* gmm  = (const float*)d_in[13];
  const float* beta = (const float*)d_in[14];
  const float* mean = (const float*)d_in[15];
  const float* var  = (const float*)d_in[16];

  const int N = in_sizes[0] / HF;
  const int E = in_sizes[1] / 2;
  const int* src = ei;
  const int* dst = ei + E;

  const size_t nf = (size_t)N * HF;
  float* ws     = (float*)d_ws;
  float* agg_h  = ws;              // agg_ea, then reused as inter-layer h buffer
  float* Cmat   = ws + nf;
  float* h1     = ws + 2 * nf;
  float* gsum   = ws + 3 * nf;
  float* deg    = ws + 4 * nf;
  float* dinv   = deg + N;
  float* bns    = dinv + N;
  float* bnb    = bns + HF;
  const size_t need = (4 * nf + 2 * (size_t)N + 2 * HF) * sizeof(float);
  if (ws_size < need) return;

  const int ntiles = (N + MTILE - 1) / MTILE;
  const long long n128 = (long long)N * HF;

  // ---- precompute (layer-invariant): deg, agg_ea, deg_inv, bn affine, Cmat
  init_nodes<<<(unsigned)((n128 + 255) / 256), 256, 0, stream>>>(agg_h, deg, n128, N);
  deg_accum<<<(E + 255) / 256, 256, 0, stream>>>(dst, deg, E);
  scatter_rows<<<(E + 7) / 8, 256, 0, stream>>>(ea, nullptr, dst, agg_h, E);
  finalize_stats<<<(N + 255) / 256, 256, 0, stream>>>(deg, dinv, N, gmm, beta, mean, var, bns, bnb);
  // Cmat = deg_inv * (agg_ea @ Wc2^T) + bc     (Wc2 = Wc[:,128:256], ldw=256)
  gemm128_wmma<<<ntiles, 256, 0, stream>>>(agg_h, Wc, 256, 128, dinv, nullptr, bc,
                                           nullptr, nullptr, 0, Cmat, nullptr, N);

  // ---- layers
  const float* h = x;
  for (int i = 0; i < 4; ++i) {
    // h1 = h @ Wi^T ; gsum = h1 (self-loop seed)
    gemm128_wmma<<<ntiles, 256, 0, stream>>>(h, Wl[i], HF, 0, nullptr, nullptr, nullptr,
                                             nullptr, nullptr, 0, h1, gsum, N);
    // gsum[dst] += h1[src]
    scatter_rows<<<(E + 7) / 8, 256, 0, stream>>>(h1, src, dst, gsum, E);
    // h = epi( deg_inv * (gsum @ Wc1^T) + Cmat + b_i )  [BN+tanh except last]
    const int last = (i == 3);
    float* out = last ? (float*)d_out : agg_h;
    gemm128_wmma<<<ntiles, 256, 0, stream>>>(gsum, Wc, 256, 0, dinv, Cmat, bl[i],
                                             last ? nullptr : bns, last ? nullptr : bnb,
                                             last ? 0 : 1, out, nullptr, N);
    h = agg_h;
  }
}